// ProjectedAdaptiveLogSoftmax_2199023256015
// MI455X (gfx1250) — compile-verified
//
#include <hip/hip_runtime.h>
#include <math.h>

// ---------------------------------------------------------------------------
// Adaptive log-softmax NLL for MI455X (gfx1250, wave32).
// GEMMs use V_WMMA_F32_16X16X4_F32 (fp32 matrix path; reference is fp32).
// ~157 GFLOP vs ~157MB unique weight bytes -> compute bound; all weights fit
// the 192MB L2, so streaming weight tiles per 32-row block stays L2-resident.
// A-blocks are staged in LDS (320KB/WGP) to avoid re-pulling the reused
// operand through the vector-memory path on every column tile.
// ---------------------------------------------------------------------------

#define NEG_BIG (-1.0e30f)

typedef __attribute__((ext_vector_type(2))) float v2f;
typedef __attribute__((ext_vector_type(8))) float v8f;

static __device__ __forceinline__ v8f wmma_f32(v2f a, v2f b, v8f c) {
  // 8 args: (neg_a, A, neg_b, B, c_mod, C, reuse_a, reuse_b)
  return __builtin_amdgcn_wmma_f32_16x16x4_f32(false, a, false, b, (short)0, c,
                                               false, false);
}

// 16-lane butterfly reductions (wave32: width-16 groups stay inside a wave)
static __device__ __forceinline__ float red16_max(float v) {
#pragma unroll
  for (int s = 8; s >= 1; s >>= 1) v = fmaxf(v, __shfl_xor(v, s, 16));
  return v;
}
static __device__ __forceinline__ float red16_sum(float v) {
#pragma unroll
  for (int s = 8; s >= 1; s >>= 1) v += __shfl_xor(v, s, 16);
  return v;
}

// ---------------------------------------------------------------------------
// Init per-(row,strip) partial (max,sum) slots.
// ---------------------------------------------------------------------------
__global__ void init_partials(float* __restrict__ partials, int n) {
  int i = blockIdx.x * blockDim.x + threadIdx.x;
  if (i < n) {
    partials[2 * i + 0] = NEG_BIG;
    partials[2 * i + 1] = 0.0f;
  }
}

// ---------------------------------------------------------------------------
// Dense GEMM: C[M,N] = A[M,K] @ B[K,N], all row-major, M%16==N%16==K%4==0.
// One wave per 16x16 output tile, 8 waves per block. (~4% of total FLOPs.)
// ---------------------------------------------------------------------------
__global__ void __launch_bounds__(256) wmma_gemm(const float* __restrict__ A,
                                                 const float* __restrict__ B,
                                                 float* __restrict__ C,
                                                 int M, int N, int K) {
  const int lane = threadIdx.x & 31;
  const int wave = threadIdx.x >> 5;
  const int l16  = lane & 15;
  const int grp  = lane >> 4;            // half-wave: selects K pair / M+8
  const int nTiles = N >> 4;
  const int mTiles = M >> 4;
  int tile = blockIdx.x * 8 + wave;
  if (tile >= mTiles * nTiles) return;   // wave-uniform exit
  const int mBase = (tile / nTiles) << 4;
  const int nBase = (tile % nTiles) << 4;

  const float* aRow = A + (size_t)(mBase + l16) * K + 2 * grp;      // A frag
  const float* bCol = B + (size_t)(2 * grp) * N + nBase + l16;      // B frag

  v8f c = {};
#pragma unroll 4
  for (int k = 0; k < K; k += 4) {
    v2f a = *(const v2f*)(aRow + k);
    v2f b;
    b.x = bCol[(size_t)k * N];
    b.y = bCol[(size_t)(k + 1) * N];
    c = wmma_f32(a, b, c);
  }

  float* cRow = C + (size_t)(mBase + 8 * grp) * N + nBase + l16;
#pragma unroll
  for (int r = 0; r < 8; ++r) cRow[(size_t)r * N] = c[r];
}

// ---------------------------------------------------------------------------
// Streaming logits + lane-local deferred logsumexp.
//   logits = A[2048,K] @ W[V,K]^T + bias
// Grid: x = row-block (32 rows), y = column strip (colsPerStrip, %64==0).
// Waves 0..3 -> rows[0..15], N-subtiles 0..3 ; waves 4..7 -> rows[16..31].
// The 32xK A-block is staged in LDS once (row pad +4 floats -> conflict-free
// 16-lane ds_load_b64). Each lane keeps a private running (max,sum) over its
// column subset; the 16-lane merge happens once at strip end.
// ---------------------------------------------------------------------------
template <int K>
__global__ void __launch_bounds__(256) wmma_lse(const float* __restrict__ A,
                                                const float* __restrict__ W,
                                                const float* __restrict__ bias,
                                                float* __restrict__ partials,
                                                int V, int colsPerStrip) {
  constexpr int LDSP = K + 4;            // padded row stride (16B-aligned)
  __shared__ float ldsA[32 * LDSP];

  const int lane = threadIdx.x & 31;
  const int wave = threadIdx.x >> 5;
  const int l16  = lane & 15;
  const int grp  = lane >> 4;
  const int msub = wave >> 2;            // 0..1
  const int nsub = wave & 3;             // 0..3
  const int mBase = blockIdx.x * 32 + msub * 16;
  const int stripStart = blockIdx.y * colsPerStrip;
  const int stripEnd   = min(stripStart + colsPerStrip, V);

  // Stage the 32-row A block into LDS (each element read from global once).
  for (int rr = wave; rr < 32; rr += 8) {
    const float* src = A + (size_t)(blockIdx.x * 32 + rr) * K;
    float* dst = ldsA + rr * LDSP;
    for (int kk = lane * 4; kk < K; kk += 128) {
      *(float4*)(dst + kk) = *(const float4*)(src + kk);
    }
  }
  __syncthreads();

  const float* aRow = ldsA + (msub * 16 + l16) * LDSP + 2 * grp;

  float lmax[8], lsum[8];
#pragma unroll
  for (int r = 0; r < 8; ++r) { lmax[r] = NEG_BIG; lsum[r] = 0.0f; }

  for (int vb = stripStart + nsub * 16; vb < stripEnd; vb += 64) {
    const int col   = vb + l16;
    const int colC  = min(col, V - 1);            // clamped (always valid)
    const float msk = (col < V) ? 1.0f : 0.0f;    // zero out pad columns
    const float* wRow = W + (size_t)colC * K + 2 * grp;

    // Prefetch the next tile's weight rows into cache (global_prefetch_b8).
    __builtin_prefetch(W + (size_t)min(col + 64, V - 1) * K + 2 * grp, 0, 1);

    v8f c = {};
#pragma unroll
    for (int k = 0; k < K; k += 4) {
      v2f a  = *(const v2f*)(aRow + k);             // ds_load_b64
      v2f bw = *(const v2f*)(wRow + k);             // global_load_b64
      v2f b;
      b.x = bw.x * msk;
      b.y = bw.y * msk;
      c = wmma_f32(a, b, c);
    }

    const float bv = (col < V) ? bias[colC] : NEG_BIG;   // pad col -> -1e30
#pragma unroll
    for (int r = 0; r < 8; ++r) {
      // lane-local online logsumexp over this lane's private column subset
      float v  = c[r] + bv;
      float nm = fmaxf(lmax[r], v);
      lsum[r]  = lsum[r] * __expf(lmax[r] - nm) + __expf(v - nm);
      lmax[r]  = nm;
    }
  }

  // Cross-lane merge (once) + write per-(row,strip) partial.
  const int p = blockIdx.y * 4 + nsub;   // partial slot (<=64 per row)
#pragma unroll
  for (int r = 0; r < 8; ++r) {
    float m = red16_max(lmax[r]);
    float s = red16_sum(lsum[r] * __expf(lmax[r] - m));
    if (l16 == 0) {
      int row = mBase + r + 8 * grp;
      float* dst = partials + ((size_t)row * 64 + p) * 2;
      dst[0] = m;
      dst[1] = s;
    }
  }
}

// ---------------------------------------------------------------------------
// Merge the <=64 per-row strip partials into lse[cluster*2048 + row].
// ---------------------------------------------------------------------------
__global__ void lse_reduce(const float* __restrict__ partials,
                           float* __restrict__ lse, int total) {
  int i = blockIdx.x * blockDim.x + threadIdx.x;
  if (i >= total) return;
  const float* p = partials + (size_t)i * 64 * 2;
  float M = NEG_BIG;
  for (int s = 0; s < 64; ++s) M = fmaxf(M, p[2 * s]);
  float S = 0.0f;
  for (int s = 0; s < 64; ++s) S += p[2 * s + 1] * __expf(p[2 * s] - M);
  lse[i] = M + logf(S);
}

// ---------------------------------------------------------------------------
// Per-token gather of the needed logits (dot products) + final NLL.
// One wave per token; branches are row-uniform within a wave.
// ---------------------------------------------------------------------------
static __device__ __forceinline__ float dot32(const float* __restrict__ a,
                                              const float* __restrict__ b,
                                              int n, int lane) {
  float s = 0.0f;
  for (int k = lane; k < n; k += 32) s += a[k] * b[k];
#pragma unroll
  for (int m = 16; m >= 1; m >>= 1) s += __shfl_xor(s, m, 32);
  return s;
}

__global__ void __launch_bounds__(256) gather_nll(
    const float* __restrict__ proj0o, const float* __restrict__ proj1o,
    const float* __restrict__ proj2o, const float* __restrict__ proj3o,
    const float* __restrict__ headW, const float* __restrict__ headB,
    const float* __restrict__ w1, const float* __restrict__ b1,
    const float* __restrict__ w2, const float* __restrict__ b2,
    const float* __restrict__ w3, const float* __restrict__ b3,
    const int* __restrict__ target, const float* __restrict__ lse,
    float* __restrict__ out, int nRows) {
  const int wave = threadIdx.x >> 5;
  const int lane = threadIdx.x & 31;
  const int row = blockIdx.x * 8 + wave;
  if (row >= nRows) return;

  const int t = target[row];
  const float hlse = lse[row];                        // head (cluster 0)
  const float* h0 = proj0o + (size_t)row * 1024;
  float nll;

  if (t < 20000) {
    float g = dot32(h0, headW + (size_t)t * 1024, 1024, lane) + headB[t];
    nll = -(g - hlse);
  } else {
    int i, l, d;
    const float* pw; const float* pb; const float* ph;
    if (t < 40000) {
      i = 1; l = 20000;  pw = w1; pb = b1; ph = proj1o + (size_t)row * 256; d = 256;
    } else if (t < 200000) {
      i = 2; l = 40000;  pw = w2; pb = b2; ph = proj2o + (size_t)row * 64;  d = 64;
    } else {
      i = 3; l = 200000; pw = w3; pb = b3; ph = proj3o + (size_t)row * 16;  d = 16;
    }
    const int col = 20003 - i;   // transformer-XL quirk: head_lp[:, H-i]
    float cl = dot32(h0, headW + (size_t)col * 1024, 1024, lane) + headB[col];
    const int ti = t - l;
    float g = dot32(ph, pw + (size_t)ti * d, d, lane) + pb[ti];
    nll = -((cl - hlse) + (g - lse[(size_t)i * nRows + row]));
  }
  if (lane == 0) out[row] = nll;
}

// ---------------------------------------------------------------------------
extern "C" void kernel_launch(void* const* d_in, const int* in_sizes, int n_in,
                              void* d_out, int out_size, void* d_ws, size_t ws_size,
                              hipStream_t stream) {
  (void)in_sizes; (void)n_in; (void)out_size; (void)ws_size;
  const float* hidden = (const float*)d_in[0];
  const int*   target = (const int*)d_in[1];
  const float* headW  = (const float*)d_in[2];
  const float* headB  = (const float*)d_in[3];
  const float* proj0  = (const float*)d_in[4];
  const float* w1 = (const float*)d_in[5];
  const float* b1 = (const float*)d_in[6];
  const float* p1 = (const float*)d_in[7];
  const float* w2 = (const float*)d_in[8];
  const float* b2 = (const float*)d_in[9];
  const float* p2 = (const float*)d_in[10];
  const float* w3 = (const float*)d_in[11];
  const float* b3 = (const float*)d_in[12];
  const float* p3 = (const float*)d_in[13];
  float* out = (float*)d_out;

  const int N = 2048, K = 1024;

  // Workspace layout (~14.6 MB of floats)
  float* ws = (float*)d_ws;
  size_t off = 0;
  float* proj0o   = ws + off; off += (size_t)N * 1024;
  float* proj1o   = ws + off; off += (size_t)N * 256;
  float* proj2o   = ws + off; off += (size_t)N * 64;
  float* proj3o   = ws + off; off += (size_t)N * 16;
  float* partials = ws + off; off += (size_t)4 * N * 64 * 2;   // [c][row][64][2]
  float* lse      = ws + off; off += (size_t)4 * N;            // [c][row]

  // 1) init partial slots (ws is poisoned to 0xAA)
  const int nPart = 4 * N * 64;
  init_partials<<<(nPart + 255) / 256, 256, 0, stream>>>(partials, nPart);

  // 2) projections: proj_c = hidden @ P_c
  auto launch_gemm = [&](const float* B, float* C, int n) {
    int tiles = (N / 16) * (n / 16);
    wmma_gemm<<<(tiles + 7) / 8, 256, 0, stream>>>(hidden, B, C, N, n, K);
  };
  launch_gemm(proj0, proj0o, 1024);
  launch_gemm(p1, proj1o, 256);
  launch_gemm(p2, proj2o, 64);
  launch_gemm(p3, proj3o, 16);

  // 3) streaming logits + online logsumexp per cluster (K is compile-time)
  auto strip_cfg = [](int V, int& S, int& cps) {
    S = (V + 2047) / 2048;
    if (S > 16) S = 16;
    if (S < 1) S = 1;
    cps = ((V + S * 64 - 1) / (S * 64)) * 64;   // multiple of 64
  };
  int S, cps;
  strip_cfg(20003, S, cps);
  wmma_lse<1024><<<dim3(N / 32, S), 256, 0, stream>>>(
      proj0o, headW, headB, partials + (size_t)0 * N * 64 * 2, 20003, cps);
  strip_cfg(20000, S, cps);
  wmma_lse<256><<<dim3(N / 32, S), 256, 0, stream>>>(
      proj1o, w1, b1, partials + (size_t)1 * N * 64 * 2, 20000, cps);
  strip_cfg(160000, S, cps);
  wmma_lse<64><<<dim3(N / 32, S), 256, 0, stream>>>(
      proj2o, w2, b2, partials + (size_t)2 * N * 64 * 2, 160000, cps);
  strip_cfg(67735, S, cps);
  wmma_lse<16><<<dim3(N / 32, S), 256, 0, stream>>>(
      proj3o, w3, b3, partials + (size_t)3 * N * 64 * 2, 67735, cps);

  // 4) merge strip partials -> lse
  lse_reduce<<<(4 * N + 255) / 256, 256, 0, stream>>>(partials, lse, 4 * N);

  // 5) gather target/cluster logits and combine
  gather_nll<<<(N + 7) / 8, 256, 0, stream>>>(proj0o, proj1o, proj2o, proj3o,
                                              headW, headB, w1, b1, w2, b2, w3, b3,
                                              target, lse, out, N);
}